// Layer_90761248899555
// MI455X (gfx1250) — compile-verified
//
#include <hip/hip_runtime.h>
#include <math.h>

#define BROWS 64
#define DK    1024
#define VSZ   128000
#define TOPP  0.9f
#define NBINS 1024

typedef float v2f __attribute__((ext_vector_type(2)));
typedef float v8f __attribute__((ext_vector_type(8)));

// ---------------------------------------------------------------------------
// Phase 1: logits = inputs @ W + b   (fp32 WMMA 16x16x4, HBM-bound on W)
// One wave owns a 64(M) x 64(N) tile -> W is streamed from HBM exactly once.
// ---------------------------------------------------------------------------
__global__ __launch_bounds__(256) void gemm_logits(const float* __restrict__ A,
                                                   const float* __restrict__ W,
                                                   const float* __restrict__ bias,
                                                   float* __restrict__ logits) {
  const int gtid = blockIdx.x * 256 + (int)threadIdx.x;
  const int wave = gtid >> 5;               // 2000 waves, 64 columns each
  const int lane = (int)threadIdx.x & 31;
  const int nl   = lane & 15;               // N (or M) index within 16-tile
  const int kh   = lane >> 4;               // K-half selector (K+0/1 vs K+2/3)
  const int n0   = wave * 64;

  const v8f vzero = {0.f, 0.f, 0.f, 0.f, 0.f, 0.f, 0.f, 0.f};
  v8f acc[4][4];
#pragma unroll
  for (int mt = 0; mt < 4; ++mt)
#pragma unroll
    for (int nt = 0; nt < 4; ++nt) acc[mt][nt] = vzero;

  for (int k = 0; k < DK; k += 4) {
    // A fragments: lane holds inputs[mt*16 + nl][k + 2*kh + {0,1}]  (L2-hot, 256 KB)
    v2f a[4];
#pragma unroll
    for (int mt = 0; mt < 4; ++mt)
      a[mt] = *(const v2f*)(A + (size_t)(mt * 16 + nl) * DK + k + 2 * kh);

    const float* wp = W + (size_t)(k + 2 * kh) * VSZ + n0 + nl;
#pragma unroll
    for (int nt = 0; nt < 4; ++nt) {
      // B fragment: lane holds W[k + 2*kh + {0,1}][n0 + nt*16 + nl] (stream once, NT)
      v2f b;
      b.x = __builtin_nontemporal_load(wp + nt * 16);
      b.y = __builtin_nontemporal_load(wp + nt * 16 + VSZ);
#pragma unroll
      for (int mt = 0; mt < 4; ++mt)
        acc[mt][nt] = __builtin_amdgcn_wmma_f32_16x16x4_f32(
            false, a[mt], false, b, (short)0, acc[mt][nt], false, false);
    }
  }

  // D layout: VGPR r holds M = r (lanes 0-15) / M = r+8 (lanes 16-31)
#pragma unroll
  for (int nt = 0; nt < 4; ++nt) {
    const int n = n0 + nt * 16 + nl;
    const float bv = bias[n];
#pragma unroll
    for (int mt = 0; mt < 4; ++mt)
#pragma unroll
      for (int r = 0; r < 8; ++r) {
        const int m = mt * 16 + 8 * kh + r;
        logits[(size_t)m * VSZ + n] = acc[mt][nt][r] + bv;
      }
  }
}

// ---------------------------------------------------------------------------
// Phase 2: per-row max and sum(exp(l - max))
// ---------------------------------------------------------------------------
__global__ __launch_bounds__(1024) void row_stats(const float* __restrict__ logits,
                                                  float* __restrict__ rowmax,
                                                  float* __restrict__ rowsum) {
  const int row = blockIdx.x;
  const int t = (int)threadIdx.x;
  const float* lr = logits + (size_t)row * VSZ;
  __shared__ float red[1024];

  float m = -3.4e38f;
  for (int i = t; i < VSZ; i += 1024) m = fmaxf(m, lr[i]);
  red[t] = m;
  __syncthreads();
  for (int s = 512; s > 0; s >>= 1) {
    if (t < s) red[t] = fmaxf(red[t], red[t + s]);
    __syncthreads();
  }
  const float rm = red[0];
  __syncthreads();

  float sum = 0.f;
  for (int i = t; i < VSZ; i += 1024) sum += __expf(lr[i] - rm);
  red[t] = sum;
  __syncthreads();
  for (int s = 512; s > 0; s >>= 1) {
    if (t < s) red[t] += red[t + s];
    __syncthreads();
  }
  if (t == 0) {
    rowmax[row] = rm;
    rowsum[row] = red[0];
  }
}

// ---------------------------------------------------------------------------
// Phase 3: top-p cutoff via log2-spaced probability histogram (LDS f32 atomics).
// Descending cumulative scan over bins == sorted-prefix cumsum at 1/16-octave
// granularity: tau = lower edge of the bin where cumsum crosses TOP_P.
// ---------------------------------------------------------------------------
__global__ __launch_bounds__(1024) void topp_threshold(const float* __restrict__ logits,
                                                       const float* __restrict__ rowmax,
                                                       const float* __restrict__ rowsum,
                                                       float* __restrict__ tau,
                                                       float* __restrict__ nucsum) {
  const int row = blockIdx.x;
  const int t = (int)threadIdx.x;
  __shared__ float h[NBINS];
  h[t] = 0.f;
  __syncthreads();

  const float rm = rowmax[row];
  const float inv = 1.f / rowsum[row];
  const float* lr = logits + (size_t)row * VSZ;
  for (int i = t; i < VSZ; i += 1024) {
    const float p = __expf(lr[i] - rm) * inv;          // (0, 1]
    float tt = -__log2f(p) * 16.0f;                    // 1/16-octave bins
    tt = fminf(fmaxf(tt, 0.0f), (float)(NBINS - 1));
    atomicAdd(&h[(int)tt], p);                         // ds_add_f32
  }
  __syncthreads();

  if (t == 0) {
    float cum = 0.f;
    int last = 0;
    for (int b = 0; b < NBINS; ++b) {                  // high-p bins first
      if (cum > TOPP) break;                           // crossing bin included;
      cum += h[b];                                     // bin of pmax (first nonempty)
      last = b;                                        // always included
    }
    tau[row] = exp2f(-(float)(last + 1) * (1.0f / 16.0f));
    nucsum[row] = (cum > 0.f) ? cum : 1.f;
  }
}

// ---------------------------------------------------------------------------
// Phase 4: Gumbel-max sample over the nucleus {p >= tau}, deterministic PCG hash.
// ---------------------------------------------------------------------------
__global__ __launch_bounds__(1024) void topp_sample(const float* __restrict__ logits,
                                                    const float* __restrict__ rowmax,
                                                    const float* __restrict__ rowsum,
                                                    const float* __restrict__ tau,
                                                    const float* __restrict__ nucsum,
                                                    int* __restrict__ out) {
  const int row = blockIdx.x;
  const int t = (int)threadIdx.x;
  __shared__ float ss[1024];
  __shared__ int si[1024];

  const float rm = rowmax[row];
  const float inv = 1.f / rowsum[row];
  const float tv = tau[row];
  const float ninv = 1.f / nucsum[row];
  const float* lr = logits + (size_t)row * VSZ;

  float best = -3.4e38f;
  int bidx = 0;
  for (int i = t; i < VSZ; i += 1024) {
    const float p = __expf(lr[i] - rm) * inv;
    if (p >= tv) {
      unsigned x = (unsigned)(row * VSZ + i) ^ (42u * 0x9E3779B9u);  // seed 42
      unsigned s = x * 747796405u + 2891336453u;                     // PCG-RXS-M-XS
      s = ((s >> ((s >> 28) + 4u)) ^ s) * 277803737u;
      s ^= s >> 22;
      const float u = ((float)s + 0.5f) * 2.3283064365386963e-10f;   // (0,1)
      const float g = -__logf(-__logf(u));                           // Gumbel(0,1)
      const float score = __logf(p * ninv) + g;
      if (score > best) { best = score; bidx = i; }
    }
  }
  ss[t] = best;
  si[t] = bidx;
  __syncthreads();
  for (int s2 = 512; s2 > 0; s2 >>= 1) {
    if (t < s2 && ss[t + s2] > ss[t]) { ss[t] = ss[t + s2]; si[t] = si[t + s2]; }
    __syncthreads();
  }
  if (t == 0) out[row] = si[0];
}

// ---------------------------------------------------------------------------
extern "C" void kernel_launch(void* const* d_in, const int* in_sizes, int n_in,
                              void* d_out, int out_size, void* d_ws, size_t ws_size,
                              hipStream_t stream) {
  (void)in_sizes; (void)n_in; (void)out_size; (void)ws_size;
  const float* inputs = (const float*)d_in[0];   // [64][1024]
  const float* W      = (const float*)d_in[1];   // [1024][128000]
  const float* bias   = (const float*)d_in[2];   // [128000]

  float* ws     = (float*)d_ws;
  float* logits = ws;                                  // 64*128000 f32 (~31.25 MB)
  float* rowmax = ws + (size_t)BROWS * VSZ;
  float* rowsum = rowmax + BROWS;
  float* tauv   = rowsum + BROWS;
  float* nsum   = tauv + BROWS;
  int*   out    = (int*)d_out;                         // [64] token ids

  const int gemm_blocks = (VSZ / 64) * 32 / 256;       // 2000 waves -> 250 blocks
  gemm_logits<<<gemm_blocks, 256, 0, stream>>>(inputs, W, bias, logits);
  row_stats<<<BROWS, 1024, 0, stream>>>(logits, rowmax, rowsum);
  topp_threshold<<<BROWS, 1024, 0, stream>>>(logits, rowmax, rowsum, tauv, nsum);
  topp_sample<<<BROWS, 1024, 0, stream>>>(logits, rowmax, rowsum, tauv, nsum, out);
}